// LightGCN_24927990186516
// MI455X (gfx1250) — compile-verified
//
#include <hip/hip_runtime.h>
#include <stdint.h>

#define TPB 256
#define WPB (TPB / 32)

// gfx1250 async copy: copies 8B/lane from the global address gaddr to the LDS address lds_addr, tracked with ASYNCcnt.
__device__ __forceinline__ void async_load_b64_to_lds(uint32_t lds_addr, const void* gaddr) {
  asm volatile("global_load_async_to_lds_b64 %0, %1, off"
               :: "v"(lds_addr), "v"(gaddr)
               : "memory");
}

// ---------------- degree count (one-time) ----------------
__global__ void lgcn_count_deg(const int* __restrict__ eu, const int* __restrict__ ei,
                               int* __restrict__ deg, int E, int U) {
  int e = blockIdx.x * TPB + threadIdx.x;
  if (e >= E) return;
  atomicAdd(&deg[eu[e]], 1);
  atomicAdd(&deg[ei[e] + U], 1);
}

// ---------------- exclusive scan -> row_ptr, cursors, d^-1/2 (one-time) ----------------
__global__ void lgcn_scan(const int* __restrict__ deg, int* __restrict__ row_ptr,
                          int* __restrict__ cursor, float* __restrict__ dis, int N) {
  __shared__ int sbuf[1024];
  __shared__ int srun;
  const int tid = threadIdx.x;
  if (tid == 0) srun = 0;
  __syncthreads();
  for (int start = 0; start < N; start += 1024) {
    int i = start + tid;
    int v = (i < N) ? deg[i] : 0;
    sbuf[tid] = v;
    __syncthreads();
    for (int off = 1; off < 1024; off <<= 1) {
      int t = (tid >= off) ? sbuf[tid - off] : 0;
      __syncthreads();
      sbuf[tid] += t;
      __syncthreads();
    }
    int incl = sbuf[tid];
    int run = srun;
    if (i < N) {
      int excl = run + incl - v;
      row_ptr[i] = excl;
      cursor[i]  = excl;
      dis[i] = rsqrtf((float)v + 1e-7f);   // (deg + 1e-7)^-0.5
    }
    __syncthreads();
    if (tid == 1023) srun = run + incl;
    __syncthreads();
  }
  if (tid == 0) row_ptr[N] = srun;
}

// ---------------- CSR fill with fused edge values (one-time) ----------------
// Stores (col, val) pairs; val = dis[u]*dis[it] is symmetric so both directions share it.
__global__ void lgcn_fill(const int* __restrict__ eu, const int* __restrict__ ei,
                          const float* __restrict__ dis, int* __restrict__ cursor,
                          int2* __restrict__ pair, int E, int U) {
  int e = blockIdx.x * TPB + threadIdx.x;
  if (e >= E) return;
  int u  = eu[e];
  int it = ei[e] + U;
  float v = dis[u] * dis[it];
  int vb = __float_as_int(v);
  int2 pu; pu.x = it; pu.y = vb;
  int2 pi; pi.x = u;  pi.y = vb;
  pair[atomicAdd(&cursor[u], 1)]  = pu;
  pair[atomicAdd(&cursor[it], 1)] = pi;
}

// ---------------- init: x0 = emb, out = 0.25*emb ----------------
__global__ void lgcn_init(const float* __restrict__ ue, const float* __restrict__ ie,
                          float* __restrict__ x0, float* __restrict__ out,
                          int U, long total) {
  long i = (long)blockIdx.x * TPB + threadIdx.x;
  if (i >= total) return;
  int r = (int)(i >> 6);
  int d = (int)(i & 63);
  float v = (r < U) ? ue[(long)r * 64 + d] : ie[(long)(r - U) * 64 + d];
  x0[i]  = v;
  out[i] = 0.25f * v;
}

// ---------------- SpMM layer: wave-per-row gather, async (col,val) double-buffer ----------------
__global__ __launch_bounds__(TPB) void lgcn_spmm(
    const int* __restrict__ row_ptr, const int2* __restrict__ pair,
    const float2* __restrict__ Xin, float2* __restrict__ Xout,
    float2* __restrict__ Out, int N) {
  __shared__ int2 s_pair[WPB][2][32];
  const int lane = threadIdx.x & 31;
  const int wid  = threadIdx.x >> 5;
  const int r = blockIdx.x * WPB + wid;
  if (r >= N) return;

  const int base = row_ptr[r];
  const int len  = row_ptr[r + 1] - base;
  float ax = 0.f, ay = 0.f;

  if (len > 0) {
    const int nch = (len + 31) >> 5;
    const uint32_t lds0 = (uint32_t)(uintptr_t)&s_pair[wid][0][lane];
    const uint32_t lds1 = (uint32_t)(uintptr_t)&s_pair[wid][1][lane];
    // prefetch chunk 0 of (col,val) pairs into LDS (async, ASYNCcnt++)
    async_load_b64_to_lds(lds0, pair + base + lane);
    for (int ch = 0; ch < nch; ++ch) {
      const int buf = ch & 1;
      if (ch + 1 < nch) {
        // overlap: fetch next pair chunk while we gather x rows of this one
        async_load_b64_to_lds(buf ? lds0 : lds1, pair + base + (ch + 1) * 32 + lane);
        __builtin_amdgcn_s_wait_asynccnt(1);   // current chunk complete (async loads in-order)
      } else {
        __builtin_amdgcn_s_wait_asynccnt(0);
      }
      asm volatile("" ::: "memory");
      const int cnt = min(32, len - (ch << 5));
      volatile const int* sp = (volatile const int*)&s_pair[wid][buf][0];
      #pragma unroll 4
      for (int k = 0; k < cnt; ++k) {
        const int   c = sp[2 * k];                    // lane-uniform ds read
        const float v = __int_as_float(sp[2 * k + 1]);
        const float2 xv = Xin[(long)c * 32 + lane];   // 8B/lane, 256B/row gather (L2)
        ax = fmaf(v, xv.x, ax);
        ay = fmaf(v, xv.y, ay);
      }
    }
  }

  const long o = (long)r * 32 + lane;
  float2 res; res.x = ax; res.y = ay;
  Xout[o] = res;                       // x_{k+1}
  float2 t = Out[o];                   // running mean accumulation (wave owns row r)
  t.x = fmaf(0.25f, ax, t.x);
  t.y = fmaf(0.25f, ay, t.y);
  Out[o] = t;
}

extern "C" void kernel_launch(void* const* d_in, const int* in_sizes, int n_in,
                              void* d_out, int out_size, void* d_ws, size_t ws_size,
                              hipStream_t stream) {
  const float* ue = (const float*)d_in[0];
  const float* ie = (const float*)d_in[1];
  const int*   eu = (const int*)d_in[2];
  const int*   ei = (const int*)d_in[3];
  const int U = in_sizes[0] / 64;
  const int I = in_sizes[1] / 64;
  const int N = U + I;
  const int E = in_sizes[2];
  (void)n_in; (void)out_size; (void)ws_size;

  char* p = (char*)d_ws;
  auto alloc = [&](size_t b) { void* q = (void*)p; p += (b + 255) & ~(size_t)255; return q; };
  int*   deg     = (int*)alloc((size_t)N * 4);
  int*   row_ptr = (int*)alloc((size_t)(N + 1) * 4);
  int*   cursor  = (int*)alloc((size_t)N * 4);
  float* dis     = (float*)alloc((size_t)N * 4);
  int2*  csr     = (int2*)alloc((size_t)2 * E * 8 + 256); // +pad: last async chunk over-read
  float* xa      = (float*)alloc((size_t)N * 64 * 4);
  float* xb      = (float*)alloc((size_t)N * 64 * 4);

  hipMemsetAsync(deg, 0, (size_t)N * 4, stream);
  const int gE = (E + TPB - 1) / TPB;
  lgcn_count_deg<<<gE, TPB, 0, stream>>>(eu, ei, deg, E, U);
  lgcn_scan<<<1, 1024, 0, stream>>>(deg, row_ptr, cursor, dis, N);
  lgcn_fill<<<gE, TPB, 0, stream>>>(eu, ei, dis, cursor, csr, E, U);

  const long total = (long)N * 64;
  const int gI = (int)((total + TPB - 1) / TPB);
  lgcn_init<<<gI, TPB, 0, stream>>>(ue, ie, xa, (float*)d_out, U, total);

  const int gS = (N + WPB - 1) / WPB;
  float2* out2 = (float2*)d_out;
  lgcn_spmm<<<gS, TPB, 0, stream>>>(row_ptr, csr, (const float2*)xa, (float2*)xb, out2, N);
  lgcn_spmm<<<gS, TPB, 0, stream>>>(row_ptr, csr, (const float2*)xb, (float2*)xa, out2, N);
  lgcn_spmm<<<gS, TPB, 0, stream>>>(row_ptr, csr, (const float2*)xa, (float2*)xb, out2, N);
}